// SepFlexConv_79156247265909
// MI455X (gfx1250) — compile-verified
//
#include <hip/hip_runtime.h>
#include <hip/hip_bf16.h>
#include <math.h>

#define KK 33
#define PP 1089       // 33*33
#define CIN 192
#define HIDC 64
#define NL 3
#define OUTC 384
#define HH 64
#define WW 64
#define NBATCH 64

typedef float v2f __attribute__((ext_vector_type(2)));
typedef float v8f __attribute__((ext_vector_type(8)));

// ---------------------------------------------------------------------------
// Kernel 1: MAGNet kernel generation -> masked depthwise kernel kern[C][33*33]
// One thread per grid point p; h vector lives in LDS (per-thread row, padded).
// ---------------------------------------------------------------------------
__global__ __launch_bounds__(64) void gen_kernel(
    const float* __restrict__ gw, const float* __restrict__ gb,
    const float* __restrict__ gg, const float* __restrict__ gm,
    const float* __restrict__ lw, const float* __restrict__ lb,
    const float* __restrict__ ow, const float* __restrict__ mmean,
    const float* __restrict__ msig, float* __restrict__ kern)
{
  __shared__ float hb[64 * 65];
  __shared__ float tb[64 * 65];
  int tid = threadIdx.x;
  int p = blockIdx.x * 64 + tid;
  bool act = p < PP;
  int pc = act ? p : 0;
  int iy = pc / KK, ix = pc - iy * KK;
  float py = -1.0f + 0.0625f * (float)iy;   // linspace(-1,1,33) step = 2/32
  float px = -1.0f + 0.0625f * (float)ix;
  float* h = hb + tid * 65;
  float* t = tb + tid * 65;

  // Gabor layer 0
  for (int u = 0; u < HIDC; ++u) {
    float s  = sinf(py * gw[u * 2 + 0] + px * gw[u * 2 + 1] + gb[u]);
    float dy = py - gm[u * 2 + 0], dx = px - gm[u * 2 + 1];
    float ey = gg[u * 2 + 0] * dy, ex = gg[u * 2 + 1] * dx;
    h[u] = s * expf(-0.5f * (ey * ey + ex * ex));
  }
  // Multiplicative filter layers
  for (int l = 0; l < NL; ++l) {
    const float* wl  = lw + l * HIDC * HIDC;
    const float* bl  = lb + l * HIDC;
    const float* gwl = gw + (l + 1) * HIDC * 2;
    const float* gbl = gb + (l + 1) * HIDC;
    const float* ggl = gg + (l + 1) * HIDC * 2;
    const float* gml = gm + (l + 1) * HIDC * 2;
    for (int o = 0; o < HIDC; ++o) {
      float acc = bl[o];
      const float* wr = wl + o * HIDC;
      #pragma unroll 8
      for (int j = 0; j < HIDC; ++j) acc += h[j] * wr[j];
      float s  = sinf(py * gwl[o * 2 + 0] + px * gwl[o * 2 + 1] + gbl[o]);
      float dy = py - gml[o * 2 + 0], dx = px - gml[o * 2 + 1];
      float ey = ggl[o * 2 + 0] * dy, ex = ggl[o * 2 + 1] * dx;
      t[o] = acc * s * expf(-0.5f * (ey * ey + ex * ex));
    }
    float* tmp = h; h = t; t = tmp;
  }
  // Gaussian mask + output-layer variance re-weighting
  float dy = (py - mmean[0]) / msig[0];
  float dx = (px - mmean[1]) / msig[1];
  float sm = expf(-0.5f * (dy * dy + dx * dx)) * rsqrtf((float)(CIN * PP));
  for (int c = 0; c < CIN; ++c) {
    float acc = 0.f;
    const float* wr = ow + c * HIDC;
    #pragma unroll 8
    for (int u = 0; u < HIDC; ++u) acc += h[u] * wr[u];
    if (act) kern[c * PP + pc] = acc * sm;
  }
}

// ---------------------------------------------------------------------------
// Kernel 2: depthwise 33x33 conv, SAME pad. Block = (batch, channel, y-tile).
// Input tile 64 rows x 96 cols in LDS (row stride 97: bank-conflict-free).
// In-range tile elements are filled with GLOBAL_LOAD_ASYNC_TO_LDS_B32 (the
// CDNA5 async-copy path, tracked by ASYNCcnt, no VGPR round trip); halo
// elements outside the image are zeroed with plain LDS stores. The flat
// pointer's low 32 bits are the wave-relative LDS byte offset (ISA aperture
// rule), which is what the instruction's VDST VGPR carries.
// Each thread makes 8 x-contiguous outputs with a rolling 8-wide window:
// 1 LDS value load + 1 broadcast weight load per 8 FMAs.
// ---------------------------------------------------------------------------
__global__ __launch_bounds__(256) void dw_kernel(
    const float* __restrict__ x, const float* __restrict__ kern,
    float* __restrict__ dwbuf, int b0)
{
  __shared__ float tile[64 * 97];
  __shared__ float sK[PP];
  int bx = blockIdx.x;
  int tIdx = bx & 1;                 // y-tile (2 of 32 rows)
  int c = (bx >> 1) % CIN;
  int bl = bx / (2 * CIN);
  int b = b0 + bl;
  int tid = threadIdx.x;
  int y0 = tIdx * 32;
  const float* xc = x + ((size_t)b * CIN + c) * (HH * WW);

  for (int i = tid; i < 64 * 96; i += 256) {
    int r = i / 96, cc = i - r * 96;
    int gy = y0 + r - 16, gx = cc - 16;
    float* dstp = &tile[r * 97 + cc];
    if ((unsigned)gy < (unsigned)HH && (unsigned)gx < (unsigned)WW) {
      const float* src = xc + gy * WW + gx;
      unsigned ldsoff = (unsigned)(uintptr_t)dstp;   // low 32 bits = LDS offset
      asm volatile("global_load_async_to_lds_b32 %0, %1, off"
                   :: "v"(ldsoff), "v"(src) : "memory");
    } else {
      *dstp = 0.f;
    }
  }
  for (int i = tid; i < PP; i += 256) sK[i] = kern[c * PP + i];
  asm volatile("s_wait_asynccnt 0x0" ::: "memory");
  __syncthreads();

  int y  = tid >> 3;                 // 0..31 output row in tile
  int xg = (tid & 7) << 3;           // 0,8,..,56 output col base
  float a0=0,a1=0,a2=0,a3=0,a4=0,a5=0,a6=0,a7=0;
  for (int ky = 0; ky < KK; ++ky) {
    const float* row = tile + (y + ky) * 97 + xg;
    const float* wr  = sK + ky * KK;
    float w0=row[0],w1=row[1],w2=row[2],w3=row[3],w4=row[4],w5=row[5],w6=row[6];
    #pragma unroll
    for (int kx = 0; kx < KK; ++kx) {
      float w7 = row[kx + 7];
      float wt = wr[kx];
      a0 = fmaf(wt, w0, a0); a1 = fmaf(wt, w1, a1);
      a2 = fmaf(wt, w2, a2); a3 = fmaf(wt, w3, a3);
      a4 = fmaf(wt, w4, a4); a5 = fmaf(wt, w5, a5);
      a6 = fmaf(wt, w6, a6); a7 = fmaf(wt, w7, a7);
      w0=w1; w1=w2; w2=w3; w3=w4; w4=w5; w5=w6; w6=w7;
    }
  }
  float* dst = dwbuf + ((size_t)bl * CIN + c) * (HH * WW) + (y0 + y) * WW + xg;
  dst[0]=a0; dst[1]=a1; dst[2]=a2; dst[3]=a3;
  dst[4]=a4; dst[5]=a5; dst[6]=a6; dst[7]=a7;
}

// ---------------------------------------------------------------------------
// Kernel 3: pointwise GEMM with fp32 WMMA (V_WMMA_F32_16X16X4_F32).
// Per batch: Out[384,4096] = PW[384,192] @ (DW[192,4096] + bias[c]).
// Block = 128x128 tile; 8 waves in 2x4; wave = 4 M-tiles x 2 N-tiles of 16x16.
// A frag (16x4 f32): lanes 0-15 K={0,1}, lanes 16-31 K={2,3}; B mirrored.
// ---------------------------------------------------------------------------
__global__ __launch_bounds__(256) void pw_kernel(
    const float* __restrict__ dwbuf, const float* __restrict__ pww,
    const float* __restrict__ bias, float* __restrict__ out, int b0)
{
  int bx = blockIdx.x;
  int mn = bx % 96;
  int bl = bx / 96;
  int bm = mn / 32;                 // 0..2  -> M block * 128
  int bn = mn - bm * 32;            // 0..31 -> N block * 128
  int tid = threadIdx.x;
  int lane = tid & 31;
  int wid = tid >> 5;
  int mw = wid & 1;                 // 2 waves in M
  int nw = wid >> 1;                // 4 waves in N
  int l16 = lane & 15;
  int kh  = (lane >> 4) << 1;       // 0 (lanes 0-15) or 2 (lanes 16-31)
  int mbase = bm * 128 + mw * 64;
  int nbase = bn * 128 + nw * 32;
  const float* dwb = dwbuf + (size_t)bl * (CIN * HH * WW);

  v8f zero = {};
  v8f acc[4][2];
  #pragma unroll
  for (int mt = 0; mt < 4; ++mt)
    #pragma unroll
    for (int nt = 0; nt < 2; ++nt) acc[mt][nt] = zero;

  for (int k0 = 0; k0 < CIN; k0 += 4) {
    int kb = k0 + kh;               // lane-half dependent K base
    v2f a[4];
    #pragma unroll
    for (int mt = 0; mt < 4; ++mt) {
      int m = mbase + mt * 16 + l16;
      a[mt] = *(const v2f*)(pww + m * CIN + kb);    // 8B-aligned (kb even)
    }
    v2f bf[2];
    #pragma unroll
    for (int nt = 0; nt < 2; ++nt) {
      int n = nbase + nt * 16 + l16;
      bf[nt][0] = dwb[(size_t)(kb    ) * (HH * WW) + n] + bias[kb];
      bf[nt][1] = dwb[(size_t)(kb + 1) * (HH * WW) + n] + bias[kb + 1];
    }
    #pragma unroll
    for (int mt = 0; mt < 4; ++mt)
      #pragma unroll
      for (int nt = 0; nt < 2; ++nt)
        acc[mt][nt] = __builtin_amdgcn_wmma_f32_16x16x4_f32(
            false, a[mt], false, bf[nt], (short)0, acc[mt][nt], false, false);
  }

  int bglob = b0 + bl;
  float* outb = out + (size_t)bglob * OUTC * (HH * WW);
  int rbase = (lane < 16) ? 0 : 8;
  #pragma unroll
  for (int mt = 0; mt < 4; ++mt) {
    #pragma unroll
    for (int nt = 0; nt < 2; ++nt) {
      int n = nbase + nt * 16 + l16;
      #pragma unroll
      for (int r = 0; r < 8; ++r) {
        int row = mbase + mt * 16 + rbase + r;
        outb[(size_t)row * (HH * WW) + n] = acc[mt][nt][r];
      }
    }
  }
}

// ---------------------------------------------------------------------------
extern "C" void kernel_launch(void* const* d_in, const int* in_sizes, int n_in,
                              void* d_out, int out_size, void* d_ws, size_t ws_size,
                              hipStream_t stream) {
  const float* x     = (const float*)d_in[0];
  const float* gw    = (const float*)d_in[1];
  const float* gb    = (const float*)d_in[2];
  const float* gg    = (const float*)d_in[3];
  const float* gm    = (const float*)d_in[4];
  const float* lw    = (const float*)d_in[5];
  const float* lb    = (const float*)d_in[6];
  const float* ow    = (const float*)d_in[7];
  const float* mmean = (const float*)d_in[8];
  const float* msig  = (const float*)d_in[9];
  const float* cbias = (const float*)d_in[10];
  const float* pww   = (const float*)d_in[11];
  float* out = (float*)d_out;

  float* kern = (float*)d_ws;
  size_t kern_bytes = (size_t)CIN * PP * sizeof(float);     // 836352 B (256-aligned)
  float* dwbuf = (float*)((char*)d_ws + kern_bytes);
  size_t per_batch = (size_t)CIN * HH * WW * sizeof(float); // 3 MiB per batch
  size_t avail = ws_size > kern_bytes ? ws_size - kern_bytes : 0;
  int NB = (int)(avail / per_batch);
  if (NB < 1) NB = 1;
  if (NB > 16) NB = 16;   // keep dw intermediate L2-resident (<=50 MB vs 192 MB L2)

  gen_kernel<<<(PP + 63) / 64, 64, 0, stream>>>(gw, gb, gg, gm, lw, lb, ow,
                                                mmean, msig, kern);
  for (int bb = 0; bb < NBATCH; bb += NB) {
    int nb = (NBATCH - bb < NB) ? (NBATCH - bb) : NB;
    dw_kernel<<<nb * 2 * CIN, 256, 0, stream>>>(x, kern, dwbuf, bb);
    pw_kernel<<<nb * 96, 256, 0, stream>>>(dwbuf, pww, cbias, out, bb);
  }
}